// MaskedLinear_558345749220
// MI455X (gfx1250) — compile-verified
//
#include <hip/hip_runtime.h>
#include <math.h>

typedef float v2f __attribute__((ext_vector_type(2)));
typedef float v8f __attribute__((ext_vector_type(8)));

#define IN_F   8192
#define OUT_F  4096
#define BATCH  128
#define INV_T  1.0e6f          // 1 / temperature
#define NSPLIT 8               // waves per block, K-split factor
#define KCHUNK (IN_F / NSPLIT) // 1024
#define TILES_M (BATCH / 16)   // 8

// ---------------- Kernel 1: per-row stable-softmax stats ----------------
// rowmax[i] = max_j mask_weight[i,j];  rowinv[i] = 1 / sum_j exp((x-max)*INV_T)
__global__ void __launch_bounds__(256)
rowstats_kernel(const float* __restrict__ mask_weight,
                float* __restrict__ rowmax,
                float* __restrict__ rowinv) {
    __shared__ float red[256];
    const int row = blockIdx.x;
    const int tid = threadIdx.x;
    const float* mrow = mask_weight + (size_t)row * OUT_F;

    float m = -INFINITY;
    for (int j = tid; j < OUT_F; j += 256) m = fmaxf(m, mrow[j]);
    red[tid] = m;
    __syncthreads();
    for (int s = 128; s > 0; s >>= 1) {
        if (tid < s) red[tid] = fmaxf(red[tid], red[tid + s]);
        __syncthreads();
    }
    const float rmax = red[0];
    __syncthreads();

    float sum = 0.0f;
    for (int j = tid; j < OUT_F; j += 256) sum += __expf((mrow[j] - rmax) * INV_T);
    red[tid] = sum;
    __syncthreads();
    for (int s = 128; s > 0; s >>= 1) {
        if (tid < s) red[tid] += red[tid + s];
        __syncthreads();
    }
    if (tid == 0) { rowmax[row] = rmax; rowinv[row] = 1.0f / red[0]; }
}

// ---------------- Kernel 2: fused masked-weight construction + WMMA GEMM ----------------
// One block per 16 output columns. 8 waves split K; each wave builds B fragments
// of `masked` on the fly (each element computed exactly once device-wide) and
// accumulates all 8 batch row-tiles with v_wmma_f32_16x16x4_f32.
__global__ void __launch_bounds__(256)
masked_gemm_wmma(const float* __restrict__ inp,
                 const float* __restrict__ weight,
                 const float* __restrict__ bias,
                 const float* __restrict__ mask_weight,
                 const float* __restrict__ mask_scale,
                 const float* __restrict__ rowmax,
                 const float* __restrict__ rowinv,
                 float* __restrict__ out) {
    __shared__ float lds[(NSPLIT / 2) * TILES_M * 8 * 32];   // 4 slots * 2048 floats = 32 KB

    const int lane  = threadIdx.x & 31;
    const int wave  = threadIdx.x >> 5;
    const int j0    = blockIdx.x * 16;
    const int n     = lane & 15;       // N (and A-row M = lane&15)
    const int khalf = lane >> 4;       // 0: K={0,1}, 1: K={2,3} within a 4-group
    const int koff  = khalf * 2;
    const int col   = j0 + n;

    const float* wrow = weight + (size_t)col * IN_F;   // weight[j, :]

    v8f acc[TILES_M] = {};

    const int kbeg = wave * KCHUNK;
    const int kend = kbeg + KCHUNK;
    for (int k0 = kbeg; k0 < kend; k0 += 4) {
        const int ka = k0 + koff;      // this lane's K pair: ka, ka+1

        // speculative prefetch of the strided mask streams (8 k-steps ahead)
        __builtin_prefetch(mask_weight + (size_t)(ka + 32) * OUT_F + col, 0, 0);
        __builtin_prefetch(mask_scale  + (size_t)(ka + 32) * OUT_F + col, 0, 0);

        // ---- build B fragment: masked[ka..ka+1][col] ----
        const v2f wpair = *(const v2f*)(wrow + ka);            // weight[j, ka], weight[j, ka+1]
        const float mw0 = mask_weight[(size_t)ka       * OUT_F + col];
        const float mw1 = mask_weight[(size_t)(ka + 1) * OUT_F + col];
        const float ms0 = mask_scale [(size_t)ka       * OUT_F + col];
        const float ms1 = mask_scale [(size_t)(ka + 1) * OUT_F + col];
        const float b0  = wpair.x * ms0 * __expf((mw0 - rowmax[ka])     * INV_T) * rowinv[ka];
        const float b1  = wpair.y * ms1 * __expf((mw1 - rowmax[ka + 1]) * INV_T) * rowinv[ka + 1];
        v2f bfrag; bfrag.x = b0; bfrag.y = b1;

        // ---- 8 A fragments (full batch) + 8 WMMAs ----
        #pragma unroll
        for (int t = 0; t < TILES_M; ++t) {
            const int m = t * 16 + n;                          // A row (M = lane&15)
            const v2f a2 = *(const v2f*)(inp + (size_t)m * IN_F + ka);
            v2f afrag; afrag.x = a2.x; afrag.y = a2.y;
            acc[t] = __builtin_amdgcn_wmma_f32_16x16x4_f32(
                false, afrag, false, bfrag, (short)0, acc[t], false, false);
        }
    }

    // ---- deterministic LDS tree reduction over the 8 K-split waves ----
    for (int stride = NSPLIT / 2; stride >= 1; stride >>= 1) {
        if (wave >= stride && wave < 2 * stride) {
            float* dst = lds + (size_t)(wave - stride) * (TILES_M * 8 * 32);
            #pragma unroll
            for (int t = 0; t < TILES_M; ++t)
                #pragma unroll
                for (int v = 0; v < 8; ++v)
                    dst[(t * 8 + v) * 32 + lane] = acc[t][v];
        }
        __syncthreads();
        if (wave < stride) {
            const float* src = lds + (size_t)wave * (TILES_M * 8 * 32);
            #pragma unroll
            for (int t = 0; t < TILES_M; ++t)
                #pragma unroll
                for (int v = 0; v < 8; ++v)
                    acc[t][v] += src[(t * 8 + v) * 32 + lane];
        }
        __syncthreads();
    }

    // ---- wave 0 adds bias and stores the 16x128 output strip ----
    if (wave == 0) {
        const float bj = bias[col];
        #pragma unroll
        for (int t = 0; t < TILES_M; ++t) {
            #pragma unroll
            for (int v = 0; v < 8; ++v) {
                const int m = t * 16 + v + khalf * 8;          // C layout: M = v (+8 for lanes 16-31)
                out[(size_t)m * OUT_F + col] = acc[t][v] + bj;
            }
        }
    }
}

extern "C" void kernel_launch(void* const* d_in, const int* in_sizes, int n_in,
                              void* d_out, int out_size, void* d_ws, size_t ws_size,
                              hipStream_t stream) {
    const float* inp         = (const float*)d_in[0];  // (128, 8192)
    const float* weight      = (const float*)d_in[1];  // (4096, 8192)
    const float* bias        = (const float*)d_in[2];  // (4096,)
    const float* mask_weight = (const float*)d_in[3];  // (8192, 4096)
    const float* mask_scale  = (const float*)d_in[4];  // (8192, 4096)
    float* out = (float*)d_out;                        // (128, 4096)

    float* rowmax = (float*)d_ws;                      // 8192 floats
    float* rowinv = rowmax + IN_F;                     // 8192 floats

    rowstats_kernel<<<IN_F, 256, 0, stream>>>(mask_weight, rowmax, rowinv);
    masked_gemm_wmma<<<OUT_F / 16, 256, 0, stream>>>(inp, weight, bias,
                                                     mask_weight, mask_scale,
                                                     rowmax, rowinv, out);
}